// Model_57827439674169
// MI455X (gfx1250) — compile-verified
//
#include <hip/hip_runtime.h>

typedef float v2f __attribute__((ext_vector_type(2)));
typedef float v8f __attribute__((ext_vector_type(8)));

#define C_CH   256
#define P_N    3
#define L_LEN  8192
#define V_N    25
#define KW     9
#define LCHUNK 128
#define WAVES  4
#define ROWS   144          // LCHUNK + 16 halo
#define LDS_STRIDE 26

__global__ __launch_bounds__(WAVES * 32)
void stgcn_fused_kernel(const float* __restrict__ x,
                        const float* __restrict__ A,
                        float* __restrict__ out) {
    __shared__ float s_lds[WAVES][ROWS * LDS_STRIDE];

    const int lane = threadIdx.x & 31;
    const int wv   = threadIdx.x >> 5;
    const int task = blockIdx.x * WAVES + wv;       // 0 .. 16383
    const int c     = task >> 6;                    // 64 chunks per channel
    const int chunk = task & 63;
    const int l0    = chunk * LCHUNK;

    const int half = lane >> 4;                     // 0 or 1
    const int n16  = lane & 15;

    // ---- preload all B tiles of A (P x 7 kchunks x 2 wtiles) into registers ----
    // B layout (4x16, KxN): lanes 0-15 -> N=lane, VGPR0 = K0 (lo half)/K2 (hi half),
    // VGPR1 = K1/K3. Out-of-range (v>=25, w>=25) entries are zero.
    v2f B[P_N][7][2];
#pragma unroll
    for (int p = 0; p < P_N; ++p)
#pragma unroll
        for (int kc = 0; kc < 7; ++kc)
#pragma unroll
            for (int wt = 0; wt < 2; ++wt) {
                const int w  = wt * 16 + n16;
                const int kk = kc * 4 + half * 2;
                const float* ap = A + p * (V_N * V_N) + kk * V_N + w;
                v2f b;
                b.x = (kk     < V_N && w < V_N) ? ap[0]   : 0.0f;
                b.y = (kk + 1 < V_N && w < V_N) ? ap[V_N] : 0.0f;
                B[p][kc][wt] = b;
            }

    float* lds = s_lds[wv];

    // Rows with l < 0 only occur for chunk 0, tile 0 -- and then the WHOLE tile
    // is zero. Zero-fill LDS rows 0..15 and skip that tile entirely, so the hot
    // loop has no bounds checks at all.
    int tstart = 0;
    if (chunk == 0) {
        tstart = 1;
        for (int i = lane; i < 16 * LDS_STRIDE; i += 32) lds[i] = 0.0f;
    }

    // ---- phase 1: s[l, w] tiles via fp32 WMMA, spilled to per-wave LDS ----
    for (int t = tstart; t < 9; ++t) {              // tiles of 16 rows, 144 rows total
        const int row_l = l0 - 16 + t * 16 + n16;   // 0 <= row_l < L_LEN guaranteed
        v8f acc0 = {0.f,0.f,0.f,0.f,0.f,0.f,0.f,0.f};
        v8f acc1 = {0.f,0.f,0.f,0.f,0.f,0.f,0.f,0.f};
#pragma unroll
        for (int p = 0; p < P_N; ++p) {
            const float* bq = x + ((size_t)(p * C_CH + c) * L_LEN + row_l) * V_N;
            const float* bp = bq + half * 2;        // element kk base for this lane
            const float  v24 = bq[24];              // always in-bounds
#pragma unroll
            for (int kc = 0; kc < 6; ++kc) {        // kk+1 <= 24: unconditional loads
                v2f a;
                a.x = bp[kc * 4];
                a.y = bp[kc * 4 + 1];
                acc0 = __builtin_amdgcn_wmma_f32_16x16x4_f32(
                           false, a, false, B[p][kc][0], (short)0, acc0, false, false);
                acc1 = __builtin_amdgcn_wmma_f32_16x16x4_f32(
                           false, a, false, B[p][kc][1], (short)0, acc1, false, false);
            }
            // kc == 6: only v=24 contributes (K lanes 1..3 are zero via B/a zeros)
            v2f a6;
            a6.x = half ? 0.0f : v24;
            a6.y = 0.0f;
            acc0 = __builtin_amdgcn_wmma_f32_16x16x4_f32(
                       false, a6, false, B[p][6][0], (short)0, acc0, false, false);
            acc1 = __builtin_amdgcn_wmma_f32_16x16x4_f32(
                       false, a6, false, B[p][6][1], (short)0, acc1, false, false);
        }
        // C/D layout: VGPR r holds M=r (lanes 0-15) and M=r+8 (lanes 16-31), N=lane&15
#pragma unroll
        for (int r = 0; r < 8; ++r) {
            const int row = t * 16 + r + half * 8;
            lds[row * LDS_STRIDE + n16] = acc0[r];
            if (n16 < V_N - 16)                     // w = 16..24 only
                lds[row * LDS_STRIDE + 16 + n16] = acc1[r];
        }
    }

    // ---- phase 2: 9-tap causal window sum along l, coalesced NT stores ----
    // out[c, l0+lo, w] = sum_{j=0..8} s[l0+lo-j, w]; lds row of (l0+lo-j) = lo+16-j
    const size_t obase = ((size_t)c * L_LEN + l0) * V_N;
    for (int it = 0; it < (LCHUNK * V_N) / 32; ++it) {   // 100 iterations
        const int idx = it * 32 + lane;
        const int lo  = idx / V_N;
        const int w   = idx - lo * V_N;
        const float* srow = lds + (lo + 8) * LDS_STRIDE + w;  // rows lo+8 .. lo+16
        float acc = 0.0f;
#pragma unroll
        for (int j = 0; j < KW; ++j) acc += srow[j * LDS_STRIDE];
        __builtin_nontemporal_store(acc, &out[obase + idx]);
    }
}

extern "C" void kernel_launch(void* const* d_in, const int* in_sizes, int n_in,
                              void* d_out, int out_size, void* d_ws, size_t ws_size,
                              hipStream_t stream) {
    const float* x = (const float*)d_in[0];   // (1, 768, 8192, 25) fp32
    const float* A = (const float*)d_in[1];   // (3, 25, 25) fp32
    float* out = (float*)d_out;               // (1, 256, 8192, 25) fp32

    const int tasks  = C_CH * (L_LEN / LCHUNK);   // 16384 wave tasks
    const int blocks = tasks / WAVES;             // 4096
    hipLaunchKernelGGL(stgcn_fused_kernel, dim3(blocks), dim3(WAVES * 32), 0, stream,
                       x, A, out);
}